// GraphConvolution_11235634446664
// MI455X (gfx1250) — compile-verified
//
#include <hip/hip_runtime.h>
#include <hip/hip_bf16.h>

typedef __attribute__((ext_vector_type(2))) float v2f;
typedef __attribute__((ext_vector_type(8))) float v8f;

#define IN_DIM   256
#define OUT_DIM  256
#define KC       32
#define NTILES   (IN_DIM / KC)        // 8 K-tiles
#define PSTRIDE  272                  // float2 stride: 2*272 % 64 == 32 -> halves on disjoint banks
#define ASTRIDE  34                   // float stride: even (b64-aligned), 16 rows on distinct banks
#define B_TILE_BYTES (16 * PSTRIDE * 8)   // 34816 B  (16 K-pairs x 256 cols, padded)
#define A_TILE_BYTES (16 * ASTRIDE * 4)   // 2176 B
#define SMEM_BYTES   (2 * B_TILE_BYTES + 2 * A_TILE_BYTES)  // 73984 B

// ---------------------------------------------------------------------------
// Kernel 0: repack w (256x256) into K-pair-interleaved layout:
//   wp[p*256 + col] = { w[2p][col], w[(2p+1)][col] }   (p = 0..127)
// This makes every WMMA B-fragment a single contiguous 8B LDS load.
// ---------------------------------------------------------------------------
__global__ void repack_w(const float* __restrict__ w, float2* __restrict__ wp) {
    int id  = blockIdx.x * 256 + threadIdx.x;   // 0 .. 32767
    int p   = id >> 8;
    int col = id & 255;
    float2 v;
    v.x = w[(2 * p)     * OUT_DIM + col];
    v.y = w[(2 * p + 1) * OUT_DIM + col];
    wp[id] = v;
}

// ---------------------------------------------------------------------------
// Kernel 1: h = x @ w  (f32 WMMA 16x16x4, double-buffered async-to-LDS tiles)
// Block = 256 threads (8 waves); block computes 16 rows x 256 cols.
// Each wave owns two 16x16 output tiles.
// ---------------------------------------------------------------------------
__global__ __launch_bounds__(256) void gemm_wmma_f32(const float* __restrict__ x,
                                                     const float2* __restrict__ wp,
                                                     float* __restrict__ h,
                                                     int nRows) {
    extern __shared__ char smem[];
    const uint32_t smemBase = (uint32_t)(uintptr_t)smem;   // low 32 bits = LDS byte offset
    const uint64_t wpBase = (uint64_t)(uintptr_t)wp;
    const uint64_t xBase  = (uint64_t)(uintptr_t)x;

    const int rowBase = blockIdx.x * 16;
    if (rowBase >= nRows) return;

    const int t    = threadIdx.x;
    const int lane = t & 31;
    const int wave = t >> 5;
    const int mn   = lane & 15;          // A row (M) == B col (N)
    const int koff = (lane >> 4) << 1;   // 0 for lanes 0-15, 2 for lanes 16-31
    const int ct0  = wave * 2;
    const int ct1  = wave * 2 + 1;

    // async-issue one K-tile (B: 8 x b128 per thread, A: 1 x b64 per thread)
    auto issue_tile = [&](int kb, int buf) {
        const uint32_t bBase = smemBase + (uint32_t)buf * B_TILE_BYTES;
        #pragma unroll
        for (int i = 0; i < 8; ++i) {
            const int cid  = i * 256 + t;          // 0..2047
            const int prow = cid >> 7;             // 0..15 (K-pair row in tile)
            const int col  = (cid & 127) * 2;      // 0,2,..,254
            const uint32_t lds  = bBase + (uint32_t)(prow * PSTRIDE + col) * 8;
            const uint32_t goff = (uint32_t)((((kb >> 1) + prow) * OUT_DIM + col) * 8);
            asm volatile("global_load_async_to_lds_b128 %0, %1, %2"
                         :: "v"(lds), "v"(goff), "s"(wpBase) : "memory");
        }
        const uint32_t aBase = smemBase + 2 * B_TILE_BYTES + (uint32_t)buf * A_TILE_BYTES;
        {
            const int r = t >> 4;                  // 0..15
            const int c = (t & 15) * 2;            // 0,2,..,30
            const uint32_t lds  = aBase + (uint32_t)(r * ASTRIDE + c) * 4;
            const uint32_t goff = (uint32_t)(((rowBase + r) * IN_DIM + kb + c) * 4);
            asm volatile("global_load_async_to_lds_b64 %0, %1, %2"
                         :: "v"(lds), "v"(goff), "s"(xBase) : "memory");
        }
    };

    v8f acc0 = {};
    v8f acc1 = {};

    issue_tile(0, 0);

    for (int ti = 0; ti < NTILES; ++ti) {
        // current tile's async writes complete (per wave), then make visible to WG
        asm volatile("s_wait_asynccnt 0x0" ::: "memory");
        __syncthreads();

        // overlap: fetch next tile into the other buffer while we compute
        if (ti + 1 < NTILES) issue_tile((ti + 1) * KC, (ti + 1) & 1);

        const int buf = ti & 1;
        const float2* Bb = (const float2*)(smem + buf * B_TILE_BYTES);
        const float*  Ab = (const float*)(smem + 2 * B_TILE_BYTES + buf * A_TILE_BYTES);

        #pragma unroll
        for (int kk = 0; kk < KC; kk += 4) {
            const int pidx = (kk + koff) >> 1;     // K-pair index (lane-dependent)

            const float2 a  = *(const float2*)(Ab + mn * ASTRIDE + kk + koff);
            const float2 b0 = Bb[pidx * PSTRIDE + ct0 * 16 + mn];
            const float2 b1 = Bb[pidx * PSTRIDE + ct1 * 16 + mn];

            v2f Af; Af.x = a.x;  Af.y = a.y;
            v2f B0; B0.x = b0.x; B0.y = b0.y;
            v2f B1; B1.x = b1.x; B1.y = b1.y;

            acc0 = __builtin_amdgcn_wmma_f32_16x16x4_f32(false, Af, false, B0,
                                                         (short)0, acc0, false, false);
            acc1 = __builtin_amdgcn_wmma_f32_16x16x4_f32(false, Af, false, B1,
                                                         (short)0, acc1, false, false);
        }
        // everyone done reading this buffer before next iteration overwrites the other
        __syncthreads();
    }

    // C/D layout: VGPR r -> rows r (lanes 0-15) and r+8 (lanes 16-31)
    const int mhi = (lane >> 4) * 8;
    #pragma unroll
    for (int r = 0; r < 8; ++r) {
        const long long row = rowBase + r + mhi;
        h[row * OUT_DIM + ct0 * 16 + mn] = acc0[r];
        h[row * OUT_DIM + ct1 * 16 + mn] = acc1[r];
    }
}

// ---------------------------------------------------------------------------
// Kernel 2: zero-init the accumulation buffer (d_out)
// ---------------------------------------------------------------------------
__global__ void zero_f4(float4* __restrict__ p, long long n4) {
    long long gid = (long long)blockIdx.x * blockDim.x + threadIdx.x;
    if (gid < n4) {
        float4 z = {0.f, 0.f, 0.f, 0.f};
        p[gid] = z;
    }
}

// ---------------------------------------------------------------------------
// Kernel 3: COO scatter-add. 32 lanes per edge, 8 cols per lane.
// h (102 MB) is L2-resident -> gather + f32 atomics are L2 traffic.
// ---------------------------------------------------------------------------
__global__ __launch_bounds__(256) void spmm_scatter(const int* __restrict__ rows,
                                                    const int* __restrict__ cols,
                                                    const float* __restrict__ vals,
                                                    const float* __restrict__ h,
                                                    float* __restrict__ out,
                                                    long long nEdges) {
    long long gid  = (long long)blockIdx.x * blockDim.x + threadIdx.x;
    long long edge = gid >> 5;
    if (edge >= nEdges) return;
    const int lane = (int)(gid & 31);

    const int   r = rows[edge];
    const int   c = cols[edge];
    const float v = vals[edge];

    const float4* hp = (const float4*)(h + (long long)c * OUT_DIM + lane * 8);
    const float4 h0 = hp[0];
    const float4 h1 = hp[1];

    float* op = out + (long long)r * OUT_DIM + lane * 8;
    unsafeAtomicAdd(op + 0, v * h0.x);
    unsafeAtomicAdd(op + 1, v * h0.y);
    unsafeAtomicAdd(op + 2, v * h0.z);
    unsafeAtomicAdd(op + 3, v * h0.w);
    unsafeAtomicAdd(op + 4, v * h1.x);
    unsafeAtomicAdd(op + 5, v * h1.y);
    unsafeAtomicAdd(op + 6, v * h1.z);
    unsafeAtomicAdd(op + 7, v * h1.w);
}

// ---------------------------------------------------------------------------
// Kernel 4: out = relu(out + b), float4 in place
// ---------------------------------------------------------------------------
__global__ void bias_relu(float4* __restrict__ out, const float4* __restrict__ b,
                          long long n4) {
    long long gid = (long long)blockIdx.x * blockDim.x + threadIdx.x;
    if (gid >= n4) return;
    const float4 bb = b[gid & 63];
    float4 v = out[gid];
    v.x = fmaxf(v.x + bb.x, 0.f);
    v.y = fmaxf(v.y + bb.y, 0.f);
    v.z = fmaxf(v.z + bb.z, 0.f);
    v.w = fmaxf(v.w + bb.w, 0.f);
    out[gid] = v;
}

// ---------------------------------------------------------------------------
extern "C" void kernel_launch(void* const* d_in, const int* in_sizes, int n_in,
                              void* d_out, int out_size, void* d_ws, size_t ws_size,
                              hipStream_t stream) {
    const float* x        = (const float*)d_in[0];
    const int*   adj_row  = (const int*)  d_in[1];
    const int*   adj_col  = (const int*)  d_in[2];
    const float* adj_vals = (const float*)d_in[3];
    const float* w        = (const float*)d_in[4];
    const float* b        = (const float*)d_in[5];
    float* out = (float*)d_out;

    const int       N = in_sizes[0] / IN_DIM;
    const long long E = in_sizes[1];

    float*  h  = (float*)d_ws;                                       // N*256 f32
    float2* wp = (float2*)((char*)d_ws + (size_t)N * OUT_DIM * 4);   // 128x256 float2

    const long long n4 = (long long)N * (OUT_DIM / 4);

    repack_w<<<(IN_DIM / 2) * OUT_DIM / 256, 256, 0, stream>>>(w, wp);

    zero_f4<<<(int)((n4 + 255) / 256), 256, 0, stream>>>((float4*)out, n4);

    gemm_wmma_f32<<<(N + 15) / 16, 256, SMEM_BYTES, stream>>>(x, wp, h, N);

    const long long tE = E * 32;
    spmm_scatter<<<(int)((tE + 255) / 256), 256, 0, stream>>>(adj_row, adj_col,
                                                              adj_vals, h, out, E);

    bias_relu<<<(int)((n4 + 255) / 256), 256, 0, stream>>>((float4*)out,
                                                           (const float4*)b, n4);
}